// Cross_Attention_75428215652947
// MI455X (gfx1250) — compile-verified
//
#include <hip/hip_runtime.h>
#include <hip/hip_bf16.h>
#include <math.h>

typedef __attribute__((ext_vector_type(2))) float v2f;
typedef __attribute__((ext_vector_type(8))) float v8f;
typedef __attribute__((ext_vector_type(4))) int   v4i;

static constexpr int BATCH = 4;
static constexpr int DIM   = 96;
static constexpr int HEADS = 2;
static constexpr int CPH   = 48;     // channels per head
static constexpr int NPIX  = 65536;  // H*W per batch

// ---------------------------------------------------------------------------
// Register-blocked fp32 GEMM, K = 96, via V_WMMA_F32_16X16X4_F32.
//   out[b][o][p] = sum_c W[b][o][c] * in[b][c][p]
// Block = 256 threads (8 waves); each wave owns a 16-pixel column tile and
// computes ALL COUT output channels for it (COUT/16 accumulators), so the
// activation matrix is read from HBM exactly once. The full weight matrix
// (COUT x 96 <= 72 KB) is staged into LDS via GLOBAL_LOAD_ASYNC_TO_LDS_B128
// (ASYNCcnt / s_wait_asynccnt) and A-fragments are served by ds_load_b64.
// ---------------------------------------------------------------------------
template <int COUT>
__global__ __launch_bounds__(256)
void wmma_gemm96_big_k(const float* __restrict__ in, long long in_bstride,
                       const float* __restrict__ W, long long w_bstride,
                       float* __restrict__ out, long long out_bstride)
{
    constexpr int NT = COUT / 16;           // # of 16-row channel tiles
    __shared__ alignas(16) float wlds[COUT * 96];

    const int lane = threadIdx.x & 31;
    const int wave = threadIdx.x >> 5;
    const int half = lane >> 4;             // which 16-lane half (K split)
    const int l    = lane & 15;
    const int b    = blockIdx.z;

    // ---- stage weights for this batch into LDS ----
    const float* Wb = W + (long long)b * w_bstride;
#if __has_builtin(__builtin_amdgcn_global_load_async_to_lds_b128)
    for (int i = threadIdx.x * 4; i < COUT * 96; i += 1024) {
        auto gp = (__attribute__((address_space(1))) v4i*)(size_t)(Wb + i);
        auto lp = (__attribute__((address_space(3))) v4i*)(unsigned)(size_t)&wlds[i];
        __builtin_amdgcn_global_load_async_to_lds_b128(gp, lp, 0, 0);
    }
#if __has_builtin(__builtin_amdgcn_s_wait_asynccnt)
    __builtin_amdgcn_s_wait_asynccnt(0);
#else
    asm volatile("s_wait_asynccnt 0" ::: "memory");
#endif
#else
    for (int i = threadIdx.x * 4; i < COUT * 96; i += 1024)
        *(float4*)&wlds[i] = *(const float4*)&Wb[i];
#endif
    __syncthreads();

    const int p0 = (blockIdx.y * 8 + wave) * 16 + l;     // pixel (N) index
    const float* inb = in + (long long)b * in_bstride + p0;

    v8f acc[NT];
    #pragma unroll
    for (int t = 0; t < NT; ++t) acc[t] = (v8f){};

    #pragma unroll 1
    for (int kk = 0; kk < 96; kk += 4) {
        // B frag: lanes 0-15 hold K rows {kk,kk+1}, lanes 16-31 {kk+2,kk+3}
        const float* xp = inb + (long long)(kk + 2 * half) * NPIX;
        v2f bb;
        bb.x = xp[0];
        bb.y = xp[NPIX];
        // 12 (or 6) WMMAs reuse this B fragment; A frags from LDS (ds_load_b64)
        #pragma unroll
        for (int t = 0; t < NT; ++t) {
            v2f a = *(const v2f*)&wlds[(t * 16 + l) * 96 + kk + 2 * half];
            acc[t] = __builtin_amdgcn_wmma_f32_16x16x4_f32(
                         false, a, false, bb, (short)0, acc[t], false, false);
        }
    }

    // D layout: VGPR r, lanes 0-15 -> M=r, lanes 16-31 -> M=r+8 ; N = l
    #pragma unroll
    for (int t = 0; t < NT; ++t) {
        float* ob = out + (long long)b * out_bstride
                        + (long long)(t * 16 + half * 8) * NPIX + p0;
        #pragma unroll
        for (int r = 0; r < 8; ++r)
            ob[(long long)r * NPIX] = acc[t][r];
    }
}

// ---------------------------------------------------------------------------
// Depthwise 3x3, same padding. One thread per output element; taps hit the
// 192 MB L2 (streaming, high spatial locality) so HBM traffic ~= compulsory.
// ---------------------------------------------------------------------------
__global__ __launch_bounds__(256)
void dwconv3_k(const float* __restrict__ in, const float* __restrict__ w,
               float* __restrict__ out, int C, long long total)
{
    long long idx = (long long)blockIdx.x * blockDim.x + threadIdx.x;
    if (idx >= total) return;
    int p = (int)(idx % NPIX);
    long long t = idx / NPIX;
    int c = (int)(t % C);
    int b = (int)(t / C);
    int yy = p >> 8, xx = p & 255;

    const float* ib = in + ((long long)b * C + c) * NPIX;
    const float* wc = w + c * 9;
    float s = 0.f;
    #pragma unroll
    for (int dy = -1; dy <= 1; ++dy) {
        int y2 = yy + dy;
        if (y2 < 0 || y2 > 255) continue;
        #pragma unroll
        for (int dx = -1; dx <= 1; ++dx) {
            int x2 = xx + dx;
            if (x2 < 0 || x2 > 255) continue;
            s += ib[y2 * 256 + x2] * wc[(dy + 1) * 3 + (dx + 1)];
        }
    }
    out[idx] = s;
}

// ---------------------------------------------------------------------------
// Reciprocal L2 row norms: rn[b*96+d] = 1/max(||row||, 1e-12).
// One 256-thread block per row, LDS tree reduction.
// ---------------------------------------------------------------------------
__global__ __launch_bounds__(256)
void rownorm_k(const float* __restrict__ base, int cbstride, float* __restrict__ rn)
{
    int b = blockIdx.x / DIM, d = blockIdx.x % DIM;
    const float* row = base + ((long long)b * cbstride + d) * NPIX;
    float s = 0.f;
    for (int i = threadIdx.x; i < NPIX; i += 256) {
        float v = row[i];
        s += v * v;
    }
    __shared__ float sm[256];
    sm[threadIdx.x] = s;
    __syncthreads();
    for (int st = 128; st > 0; st >>= 1) {
        if ((int)threadIdx.x < st) sm[threadIdx.x] += sm[threadIdx.x + st];
        __syncthreads();
    }
    if (threadIdx.x == 0)
        rn[blockIdx.x] = 1.0f / fmaxf(sqrtf(sm[0]), 1e-12f);
}

// ---------------------------------------------------------------------------
// attn[b][h][c][d] = temp[h] * rq * rk * sum_p q[b,h,c,p]*k[b,h,d,p]
// One block per (b,h, 16x16 tile of 48x48); 8 waves split K = 65536.
// Both fragments are contiguous b64 loads (reduction runs along p).
// ---------------------------------------------------------------------------
__global__ __launch_bounds__(256)
void attn_wmma_k(const float* __restrict__ q, const float* __restrict__ k,
                 const float* __restrict__ rq, const float* __restrict__ rk,
                 const float* __restrict__ temp, float* __restrict__ attn)
{
    const int lane = threadIdx.x & 31;
    const int wave = threadIdx.x >> 5;
    const int half = lane >> 4;
    const int l    = lane & 15;

    const int bid  = blockIdx.x;       // 0..71
    const int bh   = bid / 9;
    const int tile = bid % 9;
    const int ct = tile / 3, dt = tile % 3;
    const int b = bh >> 1, h = bh & 1;

    const float* qp = q + ((long long)b * 96  + h * CPH + ct * 16 + l) * NPIX; // A row
    const float* kp = k + ((long long)b * 192 + h * CPH + dt * 16 + l) * NPIX; // B col

    v8f acc = {};
    const int kchunk = NPIX / 8;
    const int k0 = wave * kchunk;
    for (int kk = k0; kk < k0 + kchunk; kk += 4) {
        v2f a  = *(const v2f*)(qp + kk + 2 * half);
        v2f bb = *(const v2f*)(kp + kk + 2 * half);
        acc = __builtin_amdgcn_wmma_f32_16x16x4_f32(
                  false, a, false, bb, (short)0, acc, false, false);
    }

    __shared__ float red[8][32][8];
    #pragma unroll
    for (int r = 0; r < 8; ++r) red[wave][lane][r] = acc[r];
    __syncthreads();

    if (wave == 0) {
        #pragma unroll
        for (int r = 0; r < 8; ++r) {
            float s = 0.f;
            #pragma unroll
            for (int w = 0; w < 8; ++w) s += red[w][lane][r];
            const int m  = r + half * 8;       // tile row
            const int cg = ct * 16 + m;        // 0..47
            const int dg = dt * 16 + l;        // 0..47
            float v = s * temp[h]
                        * rq[b * DIM + h * CPH + cg]
                        * rk[b * DIM + h * CPH + dg];
            attn[((long long)((b * 2 + h) * CPH + cg)) * CPH + dg] = v;
        }
    }
}

// ---------------------------------------------------------------------------
// Per-row top-k masked softmax, all four k values combined:
//   S = a1*sm_top24 + a2*sm_top32 + a3*sm_top36 + a4*sm_top38
// 48 values per row, 384 rows -> one thread per row.
// ---------------------------------------------------------------------------
__global__ __launch_bounds__(64)
void topk_combine_k(const float* __restrict__ attn,
                    const float* __restrict__ a1, const float* __restrict__ a2,
                    const float* __restrict__ a3, const float* __restrict__ a4,
                    float* __restrict__ S)
{
    int rid = blockIdx.x * blockDim.x + threadIdx.x;
    if (rid >= BATCH * HEADS * CPH) return;
    const float* row = attn + (long long)rid * CPH;

    float vals[CPH], srt[CPH];
    for (int i = 0; i < CPH; ++i) { vals[i] = row[i]; srt[i] = row[i]; }
    for (int i = 0; i < CPH - 1; ++i) {           // selection sort, descending
        int mx = i;
        for (int j = i + 1; j < CPH; ++j)
            if (srt[j] > srt[mx]) mx = j;
        float t = srt[i]; srt[i] = srt[mx]; srt[mx] = t;
    }

    const float maxv  = srt[0];
    const int   kc[4] = {CPH / 2, CPH * 2 / 3, CPH * 3 / 4, CPH * 4 / 5}; // 24,32,36,38
    const float al[4] = {a1[0], a2[0], a3[0], a4[0]};
    float thr[4], dinv[4];
    #pragma unroll
    for (int j = 0; j < 4; ++j) {
        thr[j] = srt[kc[j] - 1];
        float den = 0.f;
        for (int i = 0; i < kc[j]; ++i) den += __expf(srt[i] - maxv);
        dinv[j] = al[j] / den;
    }

    float* so = S + (long long)rid * CPH;
    for (int d = 0; d < CPH; ++d) {
        float e = __expf(vals[d] - maxv);
        float o = 0.f;
        #pragma unroll
        for (int j = 0; j < 4; ++j)
            o += (vals[d] >= thr[j]) ? e * dinv[j] : 0.f;
        so[d] = o;
    }
}

// ---------------------------------------------------------------------------
// M[b] = proj_w @ blockdiag(S[b,0], S[b,1])   (tiny 96x96 per batch)
// ---------------------------------------------------------------------------
__global__ __launch_bounds__(256)
void compose_M_k(const float* __restrict__ P, const float* __restrict__ S,
                 float* __restrict__ M)
{
    int idx = blockIdx.x * blockDim.x + threadIdx.x;
    if (idx >= BATCH * DIM * DIM) return;
    int c = idx % DIM;
    int t = idx / DIM;
    int o = t % DIM;
    int b = t / DIM;
    int h = c / CPH, cl = c % CPH;
    const float* Sr = S + (long long)(b * 2 + h) * CPH * CPH;
    float s = 0.f;
    #pragma unroll 8
    for (int cc = 0; cc < CPH; ++cc)
        s += P[o * DIM + h * CPH + cc] * Sr[cc * CPH + cl];
    M[((long long)b * DIM + o) * DIM + c] = s;
}

// ---------------------------------------------------------------------------
extern "C" void kernel_launch(void* const* d_in, const int* in_sizes, int n_in,
                              void* d_out, int out_size, void* d_ws, size_t ws_size,
                              hipStream_t stream)
{
    const float* x      = (const float*)d_in[0];
    const float* y      = (const float*)d_in[1];
    const float* temp   = (const float*)d_in[2];
    const float* kv_w   = (const float*)d_in[3];
    const float* kv_dww = (const float*)d_in[4];
    const float* q_w    = (const float*)d_in[5];
    const float* q_dww  = (const float*)d_in[6];
    const float* proj_w = (const float*)d_in[7];
    const float* a1     = (const float*)d_in[8];
    const float* a2     = (const float*)d_in[9];
    const float* a3     = (const float*)d_in[10];
    const float* a4     = (const float*)d_in[11];
    float* out = (float*)d_out;

    // Workspace layout (fp32 elements)
    float* ws   = (float*)d_ws;
    float* t_kv = ws;                                       // 4*192*65536  (conv1x1 kv)
    float* t_q  = t_kv + (long long)BATCH * 192 * NPIX;     // 4*96*65536   (conv1x1 q)
    float* kvd  = t_q  + (long long)BATCH * 96  * NPIX;     // 4*192*65536  (dwconv kv: k=ch0-95, v=ch96-191)
    float* dwq  = kvd  + (long long)BATCH * 192 * NPIX;     // 4*96*65536   (dwconv q)
    float* rq   = dwq  + (long long)BATCH * 96  * NPIX;     // 384
    float* rk   = rq   + BATCH * DIM;                       // 384
    float* attn = rk   + BATCH * DIM;                       // 4*2*48*48
    float* S    = attn + BATCH * HEADS * CPH * CPH;         // 4*2*48*48
    float* Mb   = S    + BATCH * HEADS * CPH * CPH;         // 4*96*96

    const long long sbx  = (long long)DIM * NPIX;       // 96-channel batch stride
    const long long sbkv = (long long)2 * DIM * NPIX;   // 192-channel batch stride

    // 1) conv1x1 kv : [192x96] @ x  -> t_kv     (WMMA, x read once)
    wmma_gemm96_big_k<192><<<dim3(1, NPIX / 128, BATCH), 256, 0, stream>>>(
        x, sbx, kv_w, 0LL, t_kv, sbkv);
    // 2) conv1x1 q  : [96x96] @ y   -> t_q      (WMMA)
    wmma_gemm96_big_k<96><<<dim3(1, NPIX / 128, BATCH), 256, 0, stream>>>(
        y, sbx, q_w, 0LL, t_q, sbx);

    // 3) depthwise 3x3
    {
        long long tot = (long long)BATCH * 192 * NPIX;
        dwconv3_k<<<(unsigned)((tot + 255) / 256), 256, 0, stream>>>(t_kv, kv_dww, kvd, 192, tot);
    }
    {
        long long tot = (long long)BATCH * 96 * NPIX;
        dwconv3_k<<<(unsigned)((tot + 255) / 256), 256, 0, stream>>>(t_q, q_dww, dwq, 96, tot);
    }

    // 4) reciprocal row norms for q and k
    rownorm_k<<<BATCH * DIM, 256, 0, stream>>>(dwq, 96, rq);
    rownorm_k<<<BATCH * DIM, 256, 0, stream>>>(kvd, 192, rk);

    // 5) attn = norm(q) @ norm(k)^T * temperature   (WMMA, split-K over 8 waves)
    attn_wmma_k<<<BATCH * HEADS * 9, 256, 0, stream>>>(dwq, kvd, rq, rk, temp, attn);

    // 6) combined top-k masked softmax S = sum_j a_j * softmax_topk_j(attn)
    topk_combine_k<<<(BATCH * HEADS * CPH + 63) / 64, 64, 0, stream>>>(
        attn, a1, a2, a3, a4, S);

    // 7) M[b] = proj_w @ blockdiag(S[b,0], S[b,1])
    compose_M_k<<<(BATCH * DIM * DIM + 255) / 256, 256, 0, stream>>>(proj_w, S, Mb);

    // 8) out = M[b] @ v[b]   (v = channels 96..191 of kvd)   (WMMA, per-batch weights)
    wmma_gemm96_big_k<96><<<dim3(1, NPIX / 128, BATCH), 256, 0, stream>>>(
        kvd + (long long)DIM * NPIX, sbkv, Mb, (long long)DIM * DIM, out, sbx);
}